// _NeuronPrimaryTransformer_25409026523687
// MI455X (gfx1250) — compile-verified
//
#include <hip/hip_runtime.h>
#include <hip/hip_bf16.h>
#include <math.h>

#define LAYERS 28
#define DIM    1024
#define NH     16
#define HD     64
#define FFN    2816
#define VOCAB  32000
#define MAXS   2048
#define NTOK   2      // B * S

typedef __attribute__((ext_vector_type(2))) float v2f;
typedef __attribute__((ext_vector_type(8))) float v8f;

// ---------------------------------------------------------------------------
// RMSNorm: one block per token. out = x * rsqrt(mean(x^2)+eps) * w
// ---------------------------------------------------------------------------
__global__ void rmsnorm_kernel(const float* __restrict__ in, const float* __restrict__ w,
                               float* __restrict__ out, int dim)
{
    const int t = blockIdx.x;
    const float* x = in + (size_t)t * dim;
    float s = 0.f;
    for (int i = threadIdx.x; i < dim; i += blockDim.x) { float v = x[i]; s += v * v; }
    for (int off = 16; off > 0; off >>= 1) s += __shfl_down(s, off, 32);
    __shared__ float lds[9];
    const int wave = threadIdx.x >> 5, lane = threadIdx.x & 31;
    if (lane == 0) lds[wave] = s;
    __syncthreads();
    if (threadIdx.x == 0) {
        float tot = 0.f;
        for (int i = 0; i < (int)(blockDim.x >> 5); ++i) tot += lds[i];
        lds[8] = rsqrtf(tot / (float)dim + 1e-5f);
    }
    __syncthreads();
    const float inv = lds[8];
    for (int i = threadIdx.x; i < dim; i += blockDim.x)
        out[(size_t)t * dim + i] = x[i] * inv * w[i];
}

// ---------------------------------------------------------------------------
// Skinny GEMM via V_WMMA_F32_16X16X4_F32:  out[MxN] (+)= X[MxK] * W[KxN]
// Compile-time M/K/N so all strides are immediates (IOFFSET-folded loads,
// single pointer bump per 32-K group). M<=16; rows M..15 zero via lane mask
// (branch-free, EXEC all-ones through WMMA). One wave per 16-column tile.
// Weights streamed exactly once; global_prefetch_b8 96 K-rows ahead.
// ---------------------------------------------------------------------------
template<bool ACCUM, int M, int K, int N>
__global__ __launch_bounds__(128) void
gemv_wmma_kernel(const float* __restrict__ X, const float* __restrict__ W,
                 float* __restrict__ out)
{
    const int lane = threadIdx.x & 31;
    const int wave = threadIdx.x >> 5;
    const int tile = blockIdx.x * (blockDim.x >> 5) + wave;
    const int nidx = lane & 15;        // column within tile / A row
    const int half = lane >> 4;        // K-pair selector per ISA layout

    const int   arow  = (nidx < M) ? nidx : 0;     // keep loads in-bounds
    const float amask = (nidx < M) ? 1.f : 0.f;    // zero-pad rows M..15
    const float* __restrict__ xr = X + arow * K + 2 * half;
    const float* __restrict__ wp = W + (size_t)(2 * half) * N + (size_t)tile * 16 + nidx;

    v8f acc = {};
    for (int k0 = 0; k0 < K; k0 += 32) {
        __builtin_prefetch(wp + (size_t)(k0 + 96) * N, 0, 1);  // global_prefetch_b8
        #pragma unroll
        for (int kk = 0; kk < 32; kk += 4) {
            const int k = k0 + kk;
            v2f a = *(const v2f*)(xr + k) * amask;   // one b64 load (8B aligned)
            v2f b;
            b.x = wp[(size_t)k * N];                 // W[k+2*half    ][col]
            b.y = wp[(size_t)k * N + N];             // W[k+2*half + 1][col]
            acc = __builtin_amdgcn_wmma_f32_16x16x4_f32(
                      false, a, false, b, (short)0, acc, false, false);
        }
    }
    // C/D layout: lanes 0-15 hold M=r in VGPR r (rows 0..7); need rows 0..M-1
    if (lane < 16) {
        const int col = tile * 16 + nidx;
        #pragma unroll
        for (int t = 0; t < M; ++t) {
            const size_t oi = (size_t)t * N + col;
            const float r = acc[t];
            out[oi] = ACCUM ? (out[oi] + r) : r;
        }
    }
}

// ---------------------------------------------------------------------------
// RoPE in-place on q and k. One thread handles a (d, d+32) pair -> race-free.
// inv_freq[j] = theta^(-j/32), cos/sin shared by both halves.
// ---------------------------------------------------------------------------
__global__ void rope_kernel(float* __restrict__ q, float* __restrict__ k,
                            const int* __restrict__ pos_ids)
{
    const int idx = blockIdx.x * blockDim.x + threadIdx.x;   // NTOK*NH*32
    if (idx >= NTOK * NH * 32) return;
    const int d = idx & 31;
    const int h = (idx >> 5) & (NH - 1);
    const int t = idx >> 9;
    const float p   = (float)pos_ids[t];
    const float inv = powf(10000.f, -((float)d) / 32.f);
    const float ang = p * inv;
    const float c = cosf(ang), s = sinf(ang);
    const size_t base = (size_t)t * DIM + (size_t)h * HD + d;
    const float q1 = q[base], q2 = q[base + 32];
    const float k1 = k[base], k2 = k[base + 32];
    q[base]      = q1 * c - q2 * s;
    q[base + 32] = q2 * c + q1 * s;
    k[base]      = k1 * c - k2 * s;
    k[base + 32] = k2 * c + k1 * s;
}

// ---------------------------------------------------------------------------
// Attention, one block per (b, h), 64 threads (= HD).
// KV cache is zero except at cache_position, so score[t] = mask[t] (+s at cp);
// we still run the exact 2048-wide softmax against the mask for fidelity.
// ---------------------------------------------------------------------------
__global__ void attn_kernel(const float* __restrict__ q, const float* __restrict__ k,
                            const float* __restrict__ v, const float* __restrict__ mask,
                            const int* __restrict__ cache_pos, float* __restrict__ o)
{
    const int b = blockIdx.x >> 4;
    const int h = blockIdx.x & (NH - 1);
    const int d = threadIdx.x;                  // 0..63
    __shared__ float red[64];
    __shared__ float sval, mval, dval;
    const size_t base = (size_t)b * DIM + (size_t)h * HD;
    const float qd = q[base + d], kd = k[base + d], vd = v[base + d];

    red[d] = qd * kd;
    __syncthreads();
    for (int off = 32; off > 0; off >>= 1) { if (d < off) red[d] += red[d + off]; __syncthreads(); }
    if (d == 0) sval = red[0] * 0.125f;         // 1/sqrt(64)
    __syncthreads();
    const float s = sval;

    const int cp = cache_pos[0];
    const float* __restrict__ mrow = mask + (size_t)b * MAXS;

    float lm = -INFINITY;
    for (int t = d; t < MAXS; t += 64) {
        const float sc = mrow[t] + (t == cp ? s : 0.f);
        lm = fmaxf(lm, sc);
    }
    red[d] = lm; __syncthreads();
    for (int off = 32; off > 0; off >>= 1) { if (d < off) red[d] = fmaxf(red[d], red[d + off]); __syncthreads(); }
    if (d == 0) mval = red[0];
    __syncthreads();
    const float m = mval;

    float ls = 0.f;
    for (int t = d; t < MAXS; t += 64) {
        const float sc = mrow[t] + (t == cp ? s : 0.f);
        ls += expf(sc - m);
    }
    red[d] = ls; __syncthreads();
    for (int off = 32; off > 0; off >>= 1) { if (d < off) red[d] += red[d + off]; __syncthreads(); }
    if (d == 0) dval = red[0];
    __syncthreads();

    const float wcp = expf(s + mrow[cp] - m) / dval;   // only non-zero V row
    o[base + d] = wcp * vd;
}

// ---------------------------------------------------------------------------
// m = silu(g) * u
// ---------------------------------------------------------------------------
__global__ void silu_mul_kernel(const float* __restrict__ g, const float* __restrict__ u,
                                float* __restrict__ out, int n)
{
    const int i = blockIdx.x * blockDim.x + threadIdx.x;
    if (i < n) {
        const float xg = g[i];
        out[i] = (xg / (1.f + expf(-xg))) * u[i];
    }
}

// ---------------------------------------------------------------------------
extern "C" void kernel_launch(void* const* d_in, const int* in_sizes, int n_in,
                              void* d_out, int out_size, void* d_ws, size_t ws_size,
                              hipStream_t stream)
{
    const float* embeds   = (const float*)d_in[0];
    const int*   pos_ids  = (const int*)  d_in[1];
    const int*   cachepos = (const int*)  d_in[2];
    const float* mask     = (const float*)d_in[3];
    const float* wq  = (const float*)d_in[4];
    const float* wk  = (const float*)d_in[5];
    const float* wv  = (const float*)d_in[6];
    const float* wo  = (const float*)d_in[7];
    const float* wg  = (const float*)d_in[8];
    const float* wu  = (const float*)d_in[9];
    const float* wd  = (const float*)d_in[10];
    const float* ln1 = (const float*)d_in[11];
    const float* ln2 = (const float*)d_in[12];
    const float* nw  = (const float*)d_in[13];
    const float* lmw = (const float*)d_in[14];

    float* ws = (float*)d_ws;
    float* h  = ws;                    // NTOK*DIM
    float* x  = h  + NTOK * DIM;       // NTOK*DIM
    float* qb = x  + NTOK * DIM;
    float* kb = qb + NTOK * DIM;
    float* vb = kb + NTOK * DIM;
    float* ob = vb + NTOK * DIM;
    float* gb = ob + NTOK * DIM;       // NTOK*FFN
    float* ub = gb + NTOK * FFN;
    float* mb = ub + NTOK * FFN;

    hipMemcpyAsync(h, embeds, sizeof(float) * NTOK * DIM, hipMemcpyDeviceToDevice, stream);

    const int WPB = 4;                 // waves per block
    const dim3 gvBlock(32 * WPB);
    const dim3 gridDIM((DIM   / 16) / WPB);    // 16 blocks
    const dim3 gridFFN((FFN   / 16) / WPB);    // 44 blocks
    const dim3 gridVOC((VOCAB / 16) / WPB);    // 500 blocks

    for (int l = 0; l < LAYERS; ++l) {
        const float* Wq = wq + (size_t)l * DIM * DIM;
        const float* Wk = wk + (size_t)l * DIM * DIM;
        const float* Wv = wv + (size_t)l * DIM * DIM;
        const float* Wo = wo + (size_t)l * DIM * DIM;
        const float* Wg = wg + (size_t)l * DIM * FFN;
        const float* Wu = wu + (size_t)l * DIM * FFN;
        const float* Wd = wd + (size_t)l * FFN * DIM;
        const float* g1 = ln1 + (size_t)l * DIM;
        const float* g2 = ln2 + (size_t)l * DIM;

        rmsnorm_kernel<<<NTOK, 256, 0, stream>>>(h, g1, x, DIM);
        gemv_wmma_kernel<false, NTOK, DIM, DIM><<<gridDIM, gvBlock, 0, stream>>>(x, Wq, qb);
        gemv_wmma_kernel<false, NTOK, DIM, DIM><<<gridDIM, gvBlock, 0, stream>>>(x, Wk, kb);
        gemv_wmma_kernel<false, NTOK, DIM, DIM><<<gridDIM, gvBlock, 0, stream>>>(x, Wv, vb);
        rope_kernel<<<(NTOK * NH * 32 + 255) / 256, 256, 0, stream>>>(qb, kb, pos_ids);
        attn_kernel<<<NTOK * NH, 64, 0, stream>>>(qb, kb, vb, mask, cachepos, ob);
        gemv_wmma_kernel<true,  NTOK, DIM, DIM><<<gridDIM, gvBlock, 0, stream>>>(ob, Wo, h);

        rmsnorm_kernel<<<NTOK, 256, 0, stream>>>(h, g2, x, DIM);
        gemv_wmma_kernel<false, NTOK, DIM, FFN><<<gridFFN, gvBlock, 0, stream>>>(x, Wg, gb);
        gemv_wmma_kernel<false, NTOK, DIM, FFN><<<gridFFN, gvBlock, 0, stream>>>(x, Wu, ub);
        silu_mul_kernel<<<(NTOK * FFN + 255) / 256, 256, 0, stream>>>(gb, ub, mb, NTOK * FFN);
        gemv_wmma_kernel<true,  NTOK, FFN, DIM><<<gridDIM, gvBlock, 0, stream>>>(mb, Wd, h);
    }

    float* out_h      = (float*)d_out;           // (B,S,DIM) normed hidden
    float* out_logits = out_h + NTOK * DIM;      // (B,S,VOCAB)
    rmsnorm_kernel<<<NTOK, 256, 0, stream>>>(h, nw, out_h, DIM);
    gemv_wmma_kernel<false, NTOK, DIM, VOCAB><<<gridVOC, gvBlock, 0, stream>>>(out_h, lmw,
                                                                               out_logits);
}